// Linear_7284264534052
// MI455X (gfx1250) — compile-verified
//
#include <hip/hip_runtime.h>

typedef float v2f __attribute__((ext_vector_type(2)));
typedef float v4f __attribute__((ext_vector_type(4)));
typedef float v8f __attribute__((ext_vector_type(8)));

// ---------------------------------------------------------------------------
// fc_out[16,3] = concat_gap[16,448] @ weight[3,448]^T + bias[3]
// Single wave32, V_WMMA_F32_16X16X4_F32, K-loop of 112 iterations.
// A (16x4 f32): lane l -> M = l&15, K = (l>>4)*2 + {0,1}  (v2f load, 8B aligned)
// B (4x16 f32): lane l -> N = l&15, K = (l>>4)*2 + {0,1}  = weight[n][k]
// C/D (16x16 f32): VGPR r, lane l -> (M = r + 8*(l>>4), N = l&15)
// ---------------------------------------------------------------------------
__global__ __launch_bounds__(32)
void fc_wmma_kernel(const float* __restrict__ A,     // [16,448]
                    const float* __restrict__ W,     // [3,448]
                    const float* __restrict__ bias,  // [3]
                    float* __restrict__ out) {       // [16,3]
  const int lane = threadIdx.x & 31;
  const int half = lane >> 4;          // 0 or 1
  const int l15  = lane & 15;
  const int kHi  = half * 2;           // K pair base within each 4-wide step
  // Branchless zero-mask for B columns n >= 3 (EXEC must stay all-1s for WMMA)
  const float msk = (l15 < 3) ? 1.0f : 0.0f;
  const int   nr  = (l15 < 3) ? l15 : 0;

  const float* arow = A + l15 * 448 + kHi;
  const float* wrow = W + nr  * 448 + kHi;

  v8f acc = {};
#pragma unroll 4
  for (int k0 = 0; k0 < 448; k0 += 4) {
    v2f a = *(const v2f*)(arow + k0);
    v2f b = *(const v2f*)(wrow + k0);
    b.x *= msk;
    b.y *= msk;
    // 8 args: (neg_a, A, neg_b, B, c_mod, C, reuse_a, reuse_b)
    acc = __builtin_amdgcn_wmma_f32_16x16x4_f32(
        false, a, false, b, (short)0, acc, false, false);
  }

  if (l15 < 3) {
    const float bc = bias[l15];
#pragma unroll
    for (int r = 0; r < 8; ++r) {
      out[(r + half * 8) * 3 + l15] = acc[r] + bc;
    }
  }
}

// ---------------------------------------------------------------------------
// vis[b, s] = sum_c src[b, c, s] * 0.5*(weight[wOff+c] + weight[448+wOff+c])
// Memory-bound streaming: each thread produces 4 consecutive spatial outputs
// via 128-bit non-temporal loads (data is touched exactly once; out1 alone
// exceeds the 192 MB L2, so NT avoids polluting it). Channel weights staged
// once in LDS (broadcast, conflict-free).
// ---------------------------------------------------------------------------
template <int C>
__global__ __launch_bounds__(256)
void vis_kernel(const float* __restrict__ src,     // [B, C, S]
                const float* __restrict__ weight,  // [3, 448]
                float* __restrict__ dst,           // [B, S]
                int S, int nQuads, int wOff) {
  __shared__ float wsh[C];
  for (int c = threadIdx.x; c < C; c += 256)
    wsh[c] = 0.5f * (weight[wOff + c] + weight[448 + wOff + c]);
  __syncthreads();

  const int idx = blockIdx.x * 256 + threadIdx.x;
  if (idx >= nQuads) return;

  const int q  = S >> 2;              // quads per batch image
  const int b  = idx / q;
  const int sp = (idx - b * q) << 2;  // spatial element offset (16B aligned)

  const float* base = src + (long long)b * C * S + sp;

  v4f acc = {0.f, 0.f, 0.f, 0.f};
#pragma unroll 8
  for (int c = 0; c < C; ++c) {
    v4f x = __builtin_nontemporal_load((const v4f*)(base + (long long)c * S));
    acc += wsh[c] * x;
  }
  *(v4f*)(dst + (long long)b * S + sp) = acc;
}

// ---------------------------------------------------------------------------
// d_in: 0=concat_gap[16,448] 1=out1[16,64,15,64,64] 2=out2[16,128,8,32,32]
//       3=out3[16,256,4,16,16] 4=weight[3,448] 5=bias[3]
// d_out (flat, tuple order): fc[48] | vis1[983040] | vis2[131072] | vis3[16384]
// ---------------------------------------------------------------------------
extern "C" void kernel_launch(void* const* d_in, const int* in_sizes, int n_in,
                              void* d_out, int out_size, void* d_ws, size_t ws_size,
                              hipStream_t stream) {
  const float* concat_gap = (const float*)d_in[0];
  const float* out1       = (const float*)d_in[1];
  const float* out2       = (const float*)d_in[2];
  const float* out3       = (const float*)d_in[3];
  const float* weight     = (const float*)d_in[4];
  const float* bias       = (const float*)d_in[5];

  float* o  = (float*)d_out;
  float* fc = o;
  float* v1 = o + 48;
  float* v2 = v1 + 16 * 15 * 64 * 64;  // +983040
  float* v3 = v2 + 16 * 8 * 32 * 32;   // +131072

  // Tiny WMMA GEMM for fc_out
  fc_wmma_kernel<<<1, 32, 0, stream>>>(concat_gap, weight, bias, fc);

  // vis1: C=64,  S=15*64*64=61440 -> 245760 quads
  {
    const int S = 15 * 64 * 64;
    const int nQ = 16 * (S / 4);
    vis_kernel<64><<<(nQ + 255) / 256, 256, 0, stream>>>(out1, weight, v1, S, nQ, 0);
  }
  // vis2: C=128, S=8*32*32=8192 -> 32768 quads
  {
    const int S = 8 * 32 * 32;
    const int nQ = 16 * (S / 4);
    vis_kernel<128><<<(nQ + 255) / 256, 256, 0, stream>>>(out2, weight, v2, S, nQ, 64);
  }
  // vis3: C=256, S=4*16*16=1024 -> 4096 quads
  {
    const int S = 4 * 16 * 16;
    const int nQ = 16 * (S / 4);
    vis_kernel<256><<<(nQ + 255) / 256, 256, 0, stream>>>(out3, weight, v3, S, nQ, 192);
  }
}